// GameOfLifeRules_3659312136490
// MI455X (gfx1250) — compile-verified
//
#include <hip/hip_runtime.h>
#include <utility>

typedef __attribute__((ext_vector_type(2))) float v2f;
typedef __attribute__((ext_vector_type(8))) float v8f;

namespace gol {

// ---------------------------------------------------------------------------
// Compile-time enumeration of the 140 "future alive" 3x3 configurations.
// Bit k of a mask == cell k (center = bit 4), matching the reference's msel.
// s3: 84 configs with popcount 3 (padded to 96 with 0xFFFF = never-a-subset),
// s4: 56 configs with popcount 4 (alive center).
// pair(i,j)=1  <=>  s3[i] subset-of s4[j]   (|i|=3,|j|=4,|i∪j|=4 ⇔ i⊂j)
// ---------------------------------------------------------------------------
struct Tables {
  unsigned short s3[96];
  unsigned short s4[64];
};

constexpr int popc9(int m) { int c = 0; for (int k = 0; k < 9; ++k) c += (m >> k) & 1; return c; }

constexpr Tables build() {
  Tables t{};
  for (int i = 0; i < 96; ++i) t.s3[i] = 0xFFFFu;  // pad: never subset of any cfg4
  for (int i = 0; i < 64; ++i) t.s4[i] = 0u;
  int i3 = 0, i4 = 0;
  for (int m = 0; m < 512; ++m) {
    const int tot = popc9(m);
    const bool alive = ((m >> 4) & 1) != 0;
    const bool future = alive ? (tot == 3 || tot == 4) : (tot == 3);
    if (!future) continue;
    if (tot == 3) t.s3[i3++] = (unsigned short)m;   // 84 entries
    else          t.s4[i4++] = (unsigned short)m;   // 56 entries (alive, tot==4)
  }
  return t;
}

constexpr unsigned S3c(int i) { return build().s3[i]; }  // compile-time folds only
constexpr unsigned S4c(int i) { return build().s4[i]; }

// Precomputed A-operand bit masks: MT.m[row][H], bit (2*kb+j) set iff
// cfg3(row) subset-of cfg4(4*kb + j + 2*H).  row = 16t + p (pad rows -> 0).
struct MTab { unsigned m[96][2]; };

constexpr MTab buildMT() {
  MTab r{};
  constexpr Tables t = build();
  for (int row = 0; row < 96; ++row) {
    for (int H = 0; H < 2; ++H) {
      unsigned mm = 0;
      for (int idx = 0; idx < 28; ++idx) {
        const int kb = idx >> 1, j = idx & 1;
        const unsigned c3 = t.s3[row];
        const unsigned c4 = t.s4[4 * kb + j + 2 * H];
        if ((c3 & ~c4) == 0u) mm |= (1u << idx);     // pad rows (0xFFFF) never pass
      }
      r.m[row][H] = mm;
    }
  }
  return r;
}

__device__ constexpr MTab MT = buildMT();            // 768 B, runtime-indexed

// compile-time unrolled loop with true constexpr indices (keeps arrays in VGPRs)
template<typename F, int... Is>
__device__ __forceinline__ void static_for_impl(F&& f, std::integer_sequence<int, Is...>) {
  (f(std::integral_constant<int, Is>{}), ...);
}
template<int N, typename F>
__device__ __forceinline__ void static_for(F&& f) {
  static_for_impl(static_cast<F&&>(f), std::make_integer_sequence<int, N>{});
}

} // namespace gol

// ---------------------------------------------------------------------------
// One wave32 = one tile of 16 consecutive pixels (pixel = lane&15; the two
// lane halves hold complementary matrix rows of the same 16 pixels).
// prob[p] = sum_i acc_i[p] - sum_{i3} acc3[i3,p] * (P34 @ acc4)[i3,p]
// Pair GEMM on V_WMMA_F32_16X16X4_F32: 6 row-tiles x 14 K-steps = 84 WMMA.
// Fully convergent control flow: EXEC is all-1s for the whole kernel body.
// ---------------------------------------------------------------------------
__global__ __launch_bounds__(256) void gol_wmma_kernel(const float* __restrict__ x,
                                                       float* __restrict__ out,
                                                       int nTiles) {
  using namespace gol;
  const int lane = (int)(threadIdx.x & 31u);
  const int p = lane & 15;
  const int h = lane >> 4;
  const int w = (int)blockIdx.x * ((int)blockDim.x >> 5) + ((int)threadIdx.x >> 5);
  if (w >= nTiles) return;                 // wave-uniform guard (EXEC stays all-1)

  const int b  = w >> 12;                  // 4096 tiles per image (256 rows * 16 tiles)
  const int rm = w & 4095;
  const int y  = rm >> 4;
  const int tx = rm & 15;
  const int x0 = (tx << 4) + p;
  const float* img = x + (size_t)b * 65536;

  // 9 periodic-neighborhood probabilities for this lane's pixel
  float n[9];
#pragma unroll
  for (int k = 0; k < 9; ++k) {
    const int dy = k / 3 - 1, dx = k % 3 - 1;
    const int yy = (y + dy) & 255;
    const int xx = (x0 + dx) & 255;
    n[k] = img[yy * 256 + xx];
  }
  float c[9];
#pragma unroll
  for (int k = 0; k < 9; ++k) c[k] = 1.0f - n[k];

  // ---- uniform group-product tables: gA over cells 0..3, gB over cells 4..8
  float gA[16]; gA[0] = 1.0f;
  static_for<4>([&](auto C4) {
    constexpr int cell = C4.value;
    constexpr int s = 1 << cell;
    static_for<s>([&](auto M) {
      constexpr int m = s - 1 - M.value;   // descending
      gA[m + s] = gA[m] * n[cell];
      gA[m]     = gA[m] * c[cell];
    });
  });
  float gB[32]; gB[0] = 1.0f;
  static_for<5>([&](auto C5) {
    constexpr int cell = C5.value;
    constexpr int s = 1 << cell;
    static_for<s>([&](auto M) {
      constexpr int m = s - 1 - M.value;
      gB[m + s] = gB[m] * n[cell + 4];
      gB[m]     = gB[m] * c[cell + 4];
    });
  });

  // ---- per-lane acc values, convergent: compute both lane-half variants with
  // constexpr indices (register-resident gA/gB) and select on h via cndmask.
  const bool hi = (h != 0);
  float sumAcc = 0.0f;

  // S4 accs in B-operand layout: slot (2*kb+j) = acc of S4 row (4*kb+j+2h)
  float acc4[28];
  static_for<28>([&](auto I) {
    constexpr int idx = I.value;
    constexpr int kb = idx >> 1, j = idx & 1;
    constexpr unsigned m0 = S4c(4 * kb + j);
    constexpr unsigned m1 = S4c(4 * kb + j + 2);
    const float v0 = gA[m0 & 15] * gB[(m0 >> 4) & 31];
    const float v1 = gA[m1 & 15] * gB[(m1 >> 4) & 31];
    const float v = hi ? v1 : v0;
    acc4[idx] = v;
    sumAcc += v;
  });

  // S3 accs in C/D layout: slot (8*t+v) = acc of S3 row (16*t+v+8h), pads -> 0
  float acc3[48];
  static_for<48>([&](auto I) {
    constexpr int idx = I.value;
    constexpr int t = idx >> 3, v = idx & 7;
    constexpr int r0 = 16 * t + v, r1 = r0 + 8;
    float v0 = 0.0f, v1 = 0.0f;
    if constexpr (r0 < 84) { constexpr unsigned mk = S3c(r0); v0 = gA[mk & 15] * gB[(mk >> 4) & 31]; }
    if constexpr (r1 < 84) { constexpr unsigned mk = S3c(r1); v1 = gA[mk & 15] * gB[(mk >> 4) & 31]; }
    const float val = hi ? v1 : v0;
    acc3[idx] = val;
    sumAcc += val;
  });

  // ---- A-operand bit masks from the precomputed table (6 dword loads)
  unsigned mt[6];
  static_for<6>([&](auto T) {
    constexpr int t = T.value;
    mt[t] = MT.m[16 * t + p][h];
  });

  // ---- WMMA region: Y = P34(96x56) @ acc4(56x16), consumed tile-wise
  float orp = 0.0f;
#pragma unroll
  for (int t = 0; t < 6; ++t) {
    v8f C = {0.f, 0.f, 0.f, 0.f, 0.f, 0.f, 0.f, 0.f};
    const unsigned m = mt[t];
#pragma unroll
    for (int kb = 0; kb < 14; ++kb) {
      v2f A, Bv;
      A.x = (float)((m >> (2 * kb)) & 1u);
      A.y = (float)((m >> (2 * kb + 1)) & 1u);
      Bv.x = acc4[2 * kb];
      Bv.y = acc4[2 * kb + 1];
      // D = A(16x4,f32) * B(4x16,f32) + C
      C = __builtin_amdgcn_wmma_f32_16x16x4_f32(false, A, false, Bv, (short)0, C, false, false);
    }
#pragma unroll
    for (int v = 0; v < 8; ++v) orp = fmaf(acc3[8 * t + v], C[v], orp);
  }

  // combine the two lane halves (same pixel) and store
  float tot = sumAcc - orp;
  tot += __shfl_xor(tot, 16, 32);
  if (h == 0) out[(size_t)b * 65536 + (size_t)y * 256 + (size_t)x0] = tot;
}

// ---------------------------------------------------------------------------
extern "C" void kernel_launch(void* const* d_in, const int* in_sizes, int n_in,
                              void* d_out, int out_size, void* d_ws, size_t ws_size,
                              hipStream_t stream) {
  (void)n_in; (void)d_ws; (void)ws_size; (void)out_size;
  const float* x = (const float*)d_in[0];
  float* out = (float*)d_out;
  const int total = in_sizes[0];           // B*1*256*256
  const int Bn = total >> 16;              // H=W=256 fixed by the reference
  const int nTiles = Bn * 4096;            // 16-pixel tiles, one per wave32
  const int wavesPerBlock = 8;             // 256 threads
  dim3 block(256);
  dim3 grid((unsigned)((nTiles + wavesPerBlock - 1) / wavesPerBlock));
  gol_wmma_kernel<<<grid, block, 0, stream>>>(x, out, nTiles);
}